// ProtoVault_77610059039065
// MI455X (gfx1250) — compile-verified
//
#include <hip/hip_runtime.h>

// ProtoVault loss on MI455X (gfx1250).
// Reformulated onto Gram matrices so all O(B*K*D)/O(B*B*D) work is fp32 WMMA
// GEMM (C = X * Y^T), and the order-sensitive scan becomes O(B*(K+B)) scalar
// work on L2-resident matrices (total working set ~108MB < 192MB L2).

#define DIMD   2048   // FEAT_DIM
#define NPROT  512    // K
#define NB     4096   // B
#define MUC    0.9f
#define DELTAC 1.0f
#define ALPHAP 1.0f
#define ALPHAR 0.5f

typedef __attribute__((ext_vector_type(2))) float v2f;
typedef __attribute__((ext_vector_type(8))) float v8f;

// ---------------------------------------------------------------------------
// Generic C[M][N] = X[M][Dk] * Y[N][Dk]^T, fp32, V_WMMA_F32_16X16X4_F32.
// Workgroup tile 128x128, 8 waves, each wave 32x64 (2x4 accumulators).
// ---------------------------------------------------------------------------
#define TM 128
#define TN 128
#define KC 32

__global__ __launch_bounds__(256)
void gemm_xyT(const float* __restrict__ X, const float* __restrict__ Y,
              float* __restrict__ C, int M, int N, int Dk) {
    __shared__ float Xs[TM][KC + 1];
    __shared__ float Ys[TN][KC + 1];

    const int tiles_n = N / TN;
    const int bm = blockIdx.x / tiles_n;
    const int bn = blockIdx.x % tiles_n;
    const int t    = threadIdx.x;
    const int wave = t >> 5;
    const int lane = t & 31;
    const int wm = wave >> 1;         // 0..3 -> 32 rows each
    const int wn = wave & 1;          // 0..1 -> 64 cols each
    const int MB = wm * 32;
    const int NBo = wn * 64;

    v8f acc[2][4];
    const v8f zv = {0.f,0.f,0.f,0.f,0.f,0.f,0.f,0.f};
#pragma unroll
    for (int r = 0; r < 2; ++r)
#pragma unroll
        for (int c = 0; c < 4; ++c) acc[r][c] = zv;

    const float* Xb = X + (size_t)bm * TM * Dk;
    const float* Yb = Y + (size_t)bn * TN * Dk;

    const int row0 = MB + (lane & 15);
    const int col0 = NBo + (lane & 15);
    const int kb   = (lane >> 4) * 2;     // second half-wave holds K+2,K+3

    for (int k0 = 0; k0 < Dk; k0 += KC) {
        for (int idx = t; idx < TM * KC; idx += 256) {
            int r = idx >> 5;   // /32
            int c = idx & 31;
            Xs[r][c] = Xb[(size_t)r * Dk + k0 + c];
            Ys[r][c] = Yb[(size_t)r * Dk + k0 + c];
        }
        __syncthreads();
#pragma unroll
        for (int kk = 0; kk < KC; kk += 4) {
            v2f a[2], b[4];
#pragma unroll
            for (int r = 0; r < 2; ++r) {
                a[r].x = Xs[row0 + r * 16][kk + kb];
                a[r].y = Xs[row0 + r * 16][kk + kb + 1];
            }
#pragma unroll
            for (int c = 0; c < 4; ++c) {
                b[c].x = Ys[col0 + c * 16][kk + kb];
                b[c].y = Ys[col0 + c * 16][kk + kb + 1];
            }
#pragma unroll
            for (int r = 0; r < 2; ++r)
#pragma unroll
                for (int c = 0; c < 4; ++c)
                    acc[r][c] = __builtin_amdgcn_wmma_f32_16x16x4_f32(
                        false, a[r], false, b[c], (short)0, acc[r][c],
                        false, false);
        }
        __syncthreads();
    }

    // C/D layout: VGPR v: lanes0-15 -> M=v, lanes16-31 -> M=v+8; N = lane&15.
    const int mofs = (lane >> 4) * 8;
    const int ncol = lane & 15;
#pragma unroll
    for (int r = 0; r < 2; ++r)
#pragma unroll
        for (int c = 0; c < 4; ++c)
#pragma unroll
            for (int v = 0; v < 8; ++v) {
                int rr = bm * TM + MB + r * 16 + v + mofs;
                int cc = bn * TN + NBo + c * 16 + ncol;
                C[(size_t)rr * N + cc] = acc[r][c][v];
            }
}

// ---------------------------------------------------------------------------
// Sequential scan (single workgroup, 1024 threads).
// Maintains DOT = P_cur * Z^T (K-major) and pn[k] = ||p_k||^2.
// Emits assign[i] (prototype hit or -1), final per-sample weights w[i] and
// per-prototype initial coefficient c0[k] so Pf = c0*P0 + sum w_i z_i.
// ---------------------------------------------------------------------------
__global__ __launch_bounds__(1024)
void scan_kernel(const float* __restrict__ P0, const int* __restrict__ g,
                 const float* __restrict__ G, float* __restrict__ DOT,
                 int* __restrict__ assign, float* __restrict__ w,
                 float* __restrict__ c0) {
    __shared__ float pnL[NPROT];
    __shared__ float sL[NPROT];
    __shared__ int   gL[NB];     // 16KB
    __shared__ int   asgL[NB];   // 16KB
    __shared__ int   kminS;

    const int t = threadIdx.x;

    if (t < NPROT) {            // pn[k] = ||P0_k||^2
        const float* p = P0 + (size_t)t * DIMD;
        float s = 0.f;
        for (int d = 0; d < DIMD; ++d) s += p[d] * p[d];
        pnL[t] = s;
    }
    for (int i = t; i < NB; i += 1024) gL[i] = g[i];
    __syncthreads();

    for (int i = 0; i < NB; ++i) {
        if (t < NPROT) sL[t] = pnL[t] - 2.0f * DOT[(size_t)t * NB + i];
        __syncthreads();

        if (t < 32) {           // wave0 argmin over 512 (first-index ties)
            float bv = 3.4e38f;
            int   bk = 1 << 30;
            for (int u = 0; u < 16; ++u) {
                int k = t * 16 + u;         // ascending within lane
                float v = sL[k];
                if (v < bv) { bv = v; bk = k; }
            }
            for (int off = 16; off > 0; off >>= 1) {
                float ov = __shfl_xor(bv, off, 32);
                int   ok = __shfl_xor(bk, off, 32);
                if (ov < bv || (ov == bv && ok < bk)) { bv = ov; bk = ok; }
            }
            if (t == 0) {
                kminS = bk;
                if (gL[i]) {
                    float od = DOT[(size_t)bk * NB + i];
                    float zn = G[(size_t)i * NB + i];
                    pnL[bk] = MUC * MUC * pnL[bk]
                            + 2.0f * MUC * (1.0f - MUC) * od
                            + (1.0f - MUC) * (1.0f - MUC) * zn;
                    asgL[i] = bk;
                } else {
                    asgL[i] = -1;
                }
            }
        }
        __syncthreads();

        if (gL[i]) {
            const int km = kminS;
            for (int j = t; j < NB; j += 1024)
                DOT[(size_t)km * NB + j] =
                    MUC * DOT[(size_t)km * NB + j] +
                    (1.0f - MUC) * G[(size_t)i * NB + j];
            __threadfence_block();
        }
        __syncthreads();
    }

    // Backward pass: w_i = (1-mu) * mu^{#later assignments to same k}
    if (t < NPROT) {
        float m = 1.0f;
        for (int i = NB - 1; i >= 0; --i) {
            if (asgL[i] == t) { w[i] = (1.0f - MUC) * m; m *= MUC; }
        }
        c0[t] = m;
    }
    for (int i = t; i < NB; i += 1024) assign[i] = asgL[i];
}

// ---------------------------------------------------------------------------
// Pf[k] = c0[k]*P0[k] + sum_{i: assign[i]==k} w[i]*Z[i]   (one block per k)
// ---------------------------------------------------------------------------
__global__ __launch_bounds__(256)
void build_pf(const float* __restrict__ P0, const float* __restrict__ Z,
              const int* __restrict__ assign, const float* __restrict__ w,
              const float* __restrict__ c0, float* __restrict__ Pf) {
    const int k = blockIdx.x;
    __shared__ int asg[NB];
    for (int i = threadIdx.x; i < NB; i += 256) asg[i] = assign[i];
    __syncthreads();

    const int d0 = threadIdx.x;     // 8 columns per thread, stride 256
    float acc[8];
    const float ck = c0[k];
#pragma unroll
    for (int u = 0; u < 8; ++u)
        acc[u] = ck * P0[(size_t)k * DIMD + d0 + u * 256];
    for (int i = 0; i < NB; ++i) {
        if (asg[i] == k) {
            const float wi = w[i];
            const float* zr = Z + (size_t)i * DIMD;
#pragma unroll
            for (int u = 0; u < 8; ++u) acc[u] += wi * zr[d0 + u * 256];
        }
    }
#pragma unroll
    for (int u = 0; u < 8; ++u) Pf[(size_t)k * DIMD + d0 + u * 256] = acc[u];
}

__global__ void diag_pn(const float* __restrict__ PPT, float* __restrict__ pn) {
    int k = blockIdx.x * 256 + threadIdx.x;
    if (k < NPROT) pn[k] = PPT[(size_t)k * (NPROT + 1)];
}

// ---------------------------------------------------------------------------
// Per-sample terms: one wave per sample.
// pull_i = ||z_i - mean(top3 protos)||^2 via Gram identities.
// Srow[i] = sum_{j anomalous} G[i][j]  (for push loss).
// ---------------------------------------------------------------------------
__global__ __launch_bounds__(256)
void reduce1(const float* __restrict__ G, const float* __restrict__ DOTF,
             const float* __restrict__ PPT, const float* __restrict__ pn,
             const int* __restrict__ g, float* __restrict__ pullA,
             float* __restrict__ Srow) {
    const int wave = threadIdx.x >> 5;
    const int lane = threadIdx.x & 31;
    const int i = blockIdx.x * 8 + wave;
    const float* row = DOTF + (size_t)i * NPROT;   // z_i . p_k, contiguous

    int sel[3];
    int e0 = -1, e1 = -1;
#pragma unroll
    for (int p = 0; p < 3; ++p) {
        float bv = 3.4e38f;
        int   bk = 1 << 30;
        for (int u = 0; u < 16; ++u) {
            int k = lane * 16 + u;
            if (k == e0 || k == e1) continue;
            float v = pn[k] - 2.0f * row[k];
            if (v < bv) { bv = v; bk = k; }
        }
        for (int off = 16; off > 0; off >>= 1) {
            float ov = __shfl_xor(bv, off, 32);
            int   ok = __shfl_xor(bk, off, 32);
            if (ov < bv || (ov == bv && ok < bk)) { bv = ov; bk = ok; }
        }
        sel[p] = bk;
        if (p == 0) e0 = bk; else if (p == 1) e1 = bk;
    }
    const int k1 = sel[0], k2 = sel[1], k3 = sel[2];
    const float dsum  = row[k1] + row[k2] + row[k3];
    const float pnsum = pn[k1] + pn[k2] + pn[k3];
    const float cross = PPT[(size_t)k1 * NPROT + k2]
                      + PPT[(size_t)k1 * NPROT + k3]
                      + PPT[(size_t)k2 * NPROT + k3];
    const float zn = G[(size_t)i * (NB + 1)];
    const float pull = zn - (2.0f / 3.0f) * dsum + (pnsum + 2.0f * cross) / 9.0f;

    float s = 0.f;
    const float* gr = G + (size_t)i * NB;
    for (int j = lane; j < NB; j += 32)
        if (!g[j]) s += gr[j];
    for (int off = 16; off > 0; off >>= 1) s += __shfl_xor(s, off, 32);

    if (lane == 0) { pullA[i] = pull; Srow[i] = s; }
}

// ---------------------------------------------------------------------------
// Final deterministic reduction (single block) -> scalar loss.
// ---------------------------------------------------------------------------
__global__ __launch_bounds__(256)
void reduce2(const float* __restrict__ G, const int* __restrict__ g,
             const float* __restrict__ pullA, const float* __restrict__ Srow,
             float* __restrict__ out) {
    __shared__ float sh[256];
    const int t = threadIdx.x;

    auto block_sum = [&](float v) -> float {
        sh[t] = v; __syncthreads();
        for (int s = 128; s > 0; s >>= 1) {
            if (t < s) sh[t] += sh[t + s];
            __syncthreads();
        }
        float r = sh[0]; __syncthreads();
        return r;
    };

    float cnt = 0.f, anomS = 0.f, pullsum = 0.f;
    for (int i = t; i < NB; i += 256) {
        pullsum += pullA[i];
        if (!g[i]) { cnt += 1.0f; anomS += Srow[i]; }
    }
    const float cntT  = block_sum(cnt);
    const float anomT = block_sum(anomS);
    const float pullT = block_sum(pullsum);

    const float cdiv = fmaxf(cntT, 1.0f);
    const float ma2  = anomT / (cdiv * cdiv);   // ||m_a||^2

    float push = 0.f;
    for (int i = t; i < NB; i += 256) {
        float znv  = G[(size_t)i * (NB + 1)];
        float term = znv - 2.0f * Srow[i] / cdiv + ma2;
        float dist = sqrtf(fmaxf(term, 0.0f));
        push += fmaxf(DELTAC - dist, 0.0f);
    }
    const float pushT = block_sum(push);

    if (t == 0) {
        float lpull = pullT / ((float)NB * (float)DIMD);
        float lpush = (cntT > 0.f) ? (pushT / (float)NB) : 0.0f;
        out[0] = ALPHAP * lpull + ALPHAR * lpush;
    }
}

// ---------------------------------------------------------------------------
extern "C" void kernel_launch(void* const* d_in, const int* in_sizes, int n_in,
                              void* d_out, int out_size, void* d_ws,
                              size_t ws_size, hipStream_t stream) {
    (void)in_sizes; (void)n_in; (void)out_size; (void)ws_size;
    const float* Z  = (const float*)d_in[0];   // (4096, 2048)
    const int*   g  = (const int*)d_in[1];     // (4096,)
    const float* P0 = (const float*)d_in[2];   // (512, 2048)
    // d_in[3] proto_age, d_in[4] m_a: do not affect the returned loss.

    float* ws = (float*)d_ws;
    float* G     = ws;                                   // 4096*4096
    float* DOT   = G + (size_t)NB * NB;                  // 512*4096 (reused as DOTF 4096*512)
    float* Pf    = DOT + (size_t)NPROT * NB;             // 512*2048
    float* PPT   = Pf + (size_t)NPROT * DIMD;            // 512*512
    float* pn    = PPT + (size_t)NPROT * NPROT;          // 512
    float* wgt   = pn + NPROT;                           // 4096
    float* c0    = wgt + NB;                             // 512
    float* Srow  = c0 + NPROT;                           // 4096
    float* pullA = Srow + NB;                            // 4096
    int*   assign = (int*)(pullA + NB);                  // 4096

    // 1) G = Z * Z^T
    gemm_xyT<<<(NB / TM) * (NB / TN), 256, 0, stream>>>(Z, Z, G, NB, NB, DIMD);
    // 2) DOT0 = P0 * Z^T  (K-major)
    gemm_xyT<<<(NPROT / TM) * (NB / TN), 256, 0, stream>>>(P0, Z, DOT, NPROT, NB, DIMD);
    // 3) sequential EMA scan
    scan_kernel<<<1, 1024, 0, stream>>>(P0, g, G, DOT, assign, wgt, c0);
    // 4) reconstruct final prototypes
    build_pf<<<NPROT, 256, 0, stream>>>(P0, Z, assign, wgt, c0, Pf);
    // 5) PPT = Pf * Pf^T
    gemm_xyT<<<(NPROT / TM) * (NPROT / TN), 256, 0, stream>>>(Pf, Pf, PPT, NPROT, NPROT, DIMD);
    // 6) DOTF = Z * Pf^T (overwrites DOT; i-major so per-sample rows contiguous)
    gemm_xyT<<<(NB / TM) * (NPROT / TN), 256, 0, stream>>>(Z, Pf, DOT, NB, NPROT, DIMD);
    // 7) pn[k] = ||p_k||^2 from PPT diagonal
    diag_pn<<<2, 256, 0, stream>>>(PPT, pn);
    // 8) per-sample pull terms + anomaly row sums
    reduce1<<<NB / 8, 256, 0, stream>>>(G, DOT, PPT, pn, g, pullA, Srow);
    // 9) deterministic final reduction -> scalar loss
    reduce2<<<1, 256, 0, stream>>>(G, g, pullA, Srow, (float*)d_out);
}